// AdditiveUpdate_44341242364186
// MI455X (gfx1250) — compile-verified
//
#include <hip/hip_runtime.h>

#define B_ 8
#define T_ 4096
#define H_ 1024
#define M_ 2048
#define K_ 32
#define R_ 1024
#define EPS_ 1e-12f

typedef __attribute__((ext_vector_type(2))) float v2f;
typedef __attribute__((ext_vector_type(8))) float v8f;

// ---------------------------------------------------------------------------
// Kernel 1: copy encoded_input -> d_out (f32, float4 vectorized)
// ---------------------------------------------------------------------------
__global__ void __launch_bounds__(256)
copy_kernel(const float4* __restrict__ src, float4* __restrict__ dst, int n4) {
  int i = blockIdx.x * blockDim.x + threadIdx.x;
  if (i < n4) dst[i] = src[i];
}

// ---------------------------------------------------------------------------
// Kernel 2: weighted[m, r] = sum_k scores[m,k] * values[m,k,r]
// One block per mention; 256 threads; each thread owns one float4 of R=1024.
// Streams 268 MB of retrieval_values at the HBM roofline.
// ---------------------------------------------------------------------------
__global__ void __launch_bounds__(256)
weighted_kernel(const float* __restrict__ scores,
                const float4* __restrict__ values,
                float4* __restrict__ weighted) {
  __shared__ float s_sc[K_];
  const int m = blockIdx.x;
  const int tid = threadIdx.x;
  if (tid < K_) s_sc[tid] = scores[m * K_ + tid];
  __syncthreads();

  const float4* vbase = values + (size_t)m * (K_ * (R_ / 4));
  float4 acc = {0.f, 0.f, 0.f, 0.f};
#pragma unroll 4
  for (int k = 0; k < K_; ++k) {
    const float s = s_sc[k];
    const float4 v = vbase[k * (R_ / 4) + tid];
    acc.x += s * v.x;
    acc.y += s * v.y;
    acc.z += s * v.z;
    acc.w += s * v.w;
  }
  weighted[(size_t)m * (R_ / 4) + tid] = acc;
}

// ---------------------------------------------------------------------------
// Kernel 3: proj = weighted[M,R] @ W[R,H] + b, masked, atomically scattered
// into d_out at row (bp[m]*T + sp[m]).
//
// One wave computes a 32x32 output region = 2x2 tiles of 16x16 via
// V_WMMA_F32_16X16X4_F32. Register blocking reuses each A fragment across
// two N-tiles and each B fragment across two M-tiles: 4 WMMAs per 6 VMEM
// ops per k-step (vs 1 WMMA per 3 VMEM unblocked).
//
// V_WMMA_F32_16X16X4_F32 layouts (ISA 7.12.2, wave32):
//   A 16x4:  lane holds A[lane&15, 2*(lane>>4) + {0,1}]
//   B 4x16:  lane holds B[2*(lane>>4) + {0,1}, lane&15]
//   C/D:     VGPR v, lane l -> D[v + 8*(l>>4), l&15]
// ---------------------------------------------------------------------------
__global__ void __launch_bounds__(256)
gemm_scatter_kernel(const float* __restrict__ weighted,  // [M, R]
                    const float* __restrict__ Wmat,      // [R, H]
                    const float* __restrict__ bias,      // [H]
                    const int* __restrict__ bp,
                    const int* __restrict__ sp,
                    const int* __restrict__ mask,
                    float* __restrict__ out)             // [B*T, H]
{
  const int lane = threadIdx.x & 31;
  const int wave = threadIdx.x >> 5;
  const int stile = blockIdx.x * 8 + wave;          // (M/32)*(H/32) = 2048
  const int NT = H_ / 32;                           // 32
  const int tm = stile / NT;                        // 0..63
  const int tn = stile % NT;                        // 0..31
  const int mbase = tm * 32;
  const int nbase = tn * 32;

  const int rc   = lane & 15;                       // row for A, col for B
  const int half = lane >> 4;                       // K sub-pair selector

  // A fragments for the two M-tiles (8-byte aligned float2 loads)
  const float* arow0 = weighted + (size_t)(mbase + rc) * R_ + 2 * half;
  const float* arow1 = arow0 + (size_t)16 * R_;
  // B fragments for the two N-tiles
  const float* bcol0 = Wmat + (size_t)(2 * half) * H_ + (nbase + rc);
  const float* bcol1 = bcol0 + 16;

  v8f c00 = {0.f, 0.f, 0.f, 0.f, 0.f, 0.f, 0.f, 0.f};
  v8f c01 = c00, c10 = c00, c11 = c00;

#pragma unroll 2
  for (int k = 0; k < R_; k += 4) {
    const size_t ko = (size_t)k * H_;
    v2f a0 = *(const v2f*)(arow0 + k);
    v2f a1 = *(const v2f*)(arow1 + k);
    v2f b0, b1;
    b0.x = bcol0[ko];
    b0.y = bcol0[ko + H_];
    b1.x = bcol1[ko];
    b1.y = bcol1[ko + H_];
    c00 = __builtin_amdgcn_wmma_f32_16x16x4_f32(false, a0, false, b0,
                                                (short)0, c00, false, false);
    c01 = __builtin_amdgcn_wmma_f32_16x16x4_f32(false, a0, false, b1,
                                                (short)0, c01, false, false);
    c10 = __builtin_amdgcn_wmma_f32_16x16x4_f32(false, a1, false, b0,
                                                (short)0, c10, false, false);
    c11 = __builtin_amdgcn_wmma_f32_16x16x4_f32(false, a1, false, b1,
                                                (short)0, c11, false, false);
  }

  // Epilogue: bias, mask, atomic scatter into d_out
  const int n0 = nbase + rc;
  const int n1 = n0 + 16;
  const float bn0 = bias[n0];
  const float bn1 = bias[n1];

#pragma unroll
  for (int mt = 0; mt < 2; ++mt) {
    const v8f& cA = mt ? c10 : c00;                 // N-tile 0
    const v8f& cB = mt ? c11 : c01;                 // N-tile 1
    const int rbase = mbase + mt * 16 + 8 * half;   // 8 consecutive M rows
#pragma unroll
    for (int v = 0; v < 8; ++v) {
      const int m = rbase + v;
      if (mask[m] != 0) {
        const size_t row = (size_t)bp[m] * T_ + (size_t)sp[m];
        float* rp = out + row * H_;
        atomicAdd(rp + n0, cA[v] + bn0);
        atomicAdd(rp + n1, cB[v] + bn1);
      }
    }
  }
}

// ---------------------------------------------------------------------------
// Kernel 4: in-place LayerNorm over H=1024 per row of d_out.
// One 256-thread block per row; float4 I/O; wave32 shuffle reduction.
// ---------------------------------------------------------------------------
__global__ void __launch_bounds__(256)
layernorm_kernel(float* __restrict__ x,
                 const float* __restrict__ scale,
                 const float* __restrict__ bias) {
  float* rowp = x + (size_t)blockIdx.x * H_;
  const int tid = threadIdx.x;

  const float4 v = ((const float4*)rowp)[tid];
  float s  = v.x + v.y + v.z + v.w;
  float s2 = v.x * v.x + v.y * v.y + v.z * v.z + v.w * v.w;

#pragma unroll
  for (int off = 16; off > 0; off >>= 1) {
    s  += __shfl_xor(s,  off, 32);
    s2 += __shfl_xor(s2, off, 32);
  }

  __shared__ float ls[8], ls2[8];
  const int wv = tid >> 5, ln = tid & 31;
  if (ln == 0) { ls[wv] = s; ls2[wv] = s2; }
  __syncthreads();
  if (tid == 0) {
    float ts = 0.f, ts2 = 0.f;
#pragma unroll
    for (int i = 0; i < 8; ++i) { ts += ls[i]; ts2 += ls2[i]; }
    ls[0] = ts; ls2[0] = ts2;
  }
  __syncthreads();

  const float mean = ls[0] * (1.0f / H_);
  const float var  = ls2[0] * (1.0f / H_) - mean * mean;
  const float inv  = rsqrtf(var + EPS_);

  const float4 sc = ((const float4*)scale)[tid];
  const float4 bi = ((const float4*)bias)[tid];
  float4 y;
  y.x = (v.x - mean) * inv * sc.x + bi.x;
  y.y = (v.y - mean) * inv * sc.y + bi.y;
  y.z = (v.z - mean) * inv * sc.z + bi.z;
  y.w = (v.w - mean) * inv * sc.w + bi.w;
  ((float4*)rowp)[tid] = y;
}

// ---------------------------------------------------------------------------
extern "C" void kernel_launch(void* const* d_in, const int* in_sizes, int n_in,
                              void* d_out, int out_size, void* d_ws, size_t ws_size,
                              hipStream_t stream) {
  const float* encoded = (const float*)d_in[0];   // [B,T,H]
  const float* values  = (const float*)d_in[1];   // [M,K,R]
  const float* scores  = (const float*)d_in[2];   // [M,K]
  const float* Wmat    = (const float*)d_in[3];   // [R,H]
  const float* bvec    = (const float*)d_in[4];   // [H]
  const float* lnsc    = (const float*)d_in[5];   // [H]
  const float* lnbi    = (const float*)d_in[6];   // [H]
  const int*   bp      = (const int*)d_in[7];     // [M]
  const int*   sp      = (const int*)d_in[8];     // [M]
  // d_in[9] = mention_end_positions (unused, as in reference)
  const int*   mask    = (const int*)d_in[10];    // [M]

  float* out = (float*)d_out;                     // [B,T,H]
  float* weighted = (float*)d_ws;                 // [M,R] = 8 MB scratch

  // 1) out = encoded_input
  const int n4 = (B_ * T_ * H_) / 4;
  copy_kernel<<<n4 / 256, 256, 0, stream>>>((const float4*)encoded,
                                            (float4*)out, n4);

  // 2) weighted = einsum('qk,qkd')
  weighted_kernel<<<M_, 256, 0, stream>>>(scores, (const float4*)values,
                                          (float4*)weighted);

  // 3) WMMA GEMM (2x2 register-blocked) + masked atomic scatter into out
  const int stiles = (M_ / 32) * (H_ / 32);       // 2048 waves
  gemm_scatter_kernel<<<stiles / 8, 256, 0, stream>>>(weighted, Wmat, bvec,
                                                      bp, sp, mask, out);

  // 4) LayerNorm in place
  layernorm_kernel<<<B_ * T_, 256, 0, stream>>>(out, lnsc, lnbi);
}